// CostVolume_66529043415444
// MI455X (gfx1250) — compile-verified
//
#include <hip/hip_runtime.h>
#include <stdint.h>

// Cost volume: out(B,2C,D,H,W); left broadcast over d, right shifted by d.
// Pure data movement: store-bandwidth bound (403MB NT writes, ~17us floor
// at 23.3 TB/s). One b128 non-temporal store per output float4.

typedef float f4 __attribute__((ext_vector_type(4)));
typedef int   i4 __attribute__((vector_size(4 * sizeof(int))));  // builtin pointee type

#define GLOBAL_AS __attribute__((address_space(1)))
#define LDS_AS    __attribute__((address_space(3)))

__global__ __launch_bounds__(256) void cost_volume_kernel(
    const float* __restrict__ left,
    const float* __restrict__ right,
    float* __restrict__ out)
{
    constexpr int B = 4, C = 32, H = 64, W = 128, D = 48;
    constexpr int W4       = W / 4;          // 32 float4 per row
    constexpr int DSTRIDE4 = H * W4;         // 2048 float4 per d-slice
    constexpr int CSTRIDE4 = D * DSTRIDE4;   // 98304 float4 per channel

    // right row staged in LDS, padded with zeros on [W, W+64) so the
    // shifted gather rrow[4*w4 + d + k] (max index 174) never branches.
    __shared__ __align__(16) float rrow[W + 64];

    const int tid = threadIdx.x;
    const int blk = blockIdx.x;              // = (b*C + c)*H + h
    const int h   = blk & (H - 1);
    const int bc  = blk >> 6;
    const int c   = bc & (C - 1);
    const int b   = bc >> 5;

    const f4* left4  = (const f4*)left;
    const f4* right4 = (const f4*)right;
    f4*       out4   = (f4*)out;

    const int inRow4 = ((b * C + c) * H + h) * W4;

    // ---- stage right row into LDS (CDNA5 async DMA path if available) ----
#if __has_builtin(__builtin_amdgcn_global_load_async_to_lds_b128) && \
    __has_builtin(__builtin_amdgcn_s_wait_asynccnt)
    if (tid < W4) {
        // flat global address == AS1 address; low 32 bits of a flat LDS
        // pointer == AS3 byte offset (aperture lives in addr[63:32]).
        __builtin_amdgcn_global_load_async_to_lds_b128(
            (GLOBAL_AS i4*)(uintptr_t)(right4 + inRow4 + tid),
            (LDS_AS i4*)(uint32_t)(uintptr_t)&rrow[tid * 4],
            /*offset=*/0, /*cpol=*/0);
    } else if (tid < W4 + 16) {              // zero pad floats [128,192)
        f4 z = {0.0f, 0.0f, 0.0f, 0.0f};
        ((f4*)rrow)[tid] = z;
    }
    __builtin_amdgcn_s_wait_asynccnt(0);
#else
    if (tid < W4) {
        ((f4*)rrow)[tid] = right4[inRow4 + tid];
    } else if (tid < W4 + 16) {
        f4 z = {0.0f, 0.0f, 0.0f, 0.0f};
        ((f4*)rrow)[tid] = z;
    }
#endif
    __syncthreads();

    const int w4  = tid & (W4 - 1);          // loop-invariant column
    const int sub = tid >> 5;                // wave id in block, 0..7
    const f4  lv  = left4[inRow4 + w4];      // left value, reused for all d

    const int base4 = (b * 2 * C + c) * CSTRIDE4 + h * W4 + w4;

    // ---- left half: broadcast over d (6 NT b128 stores / thread) ----
#pragma unroll
    for (int it = 0; it < 6; ++it) {
        const int d = it * 8 + sub;          // covers 0..47 across waves
        __builtin_nontemporal_store(lv, &out4[base4 + d * DSTRIDE4]);
    }

    // ---- right half: shifted gather from LDS (6 NT b128 stores / thread) ----
#pragma unroll
    for (int it = 0; it < 6; ++it) {
        const int d = it * 8 + sub;
        const int s = 4 * w4 + d;
        f4 v;
        v.x = rrow[s + 0];
        v.y = rrow[s + 1];
        v.z = rrow[s + 2];
        v.w = rrow[s + 3];
        __builtin_nontemporal_store(
            v, &out4[base4 + C * CSTRIDE4 + d * DSTRIDE4]);
    }
}

extern "C" void kernel_launch(void* const* d_in, const int* in_sizes, int n_in,
                              void* d_out, int out_size, void* d_ws, size_t ws_size,
                              hipStream_t stream) {
    (void)in_sizes; (void)n_in; (void)out_size; (void)d_ws; (void)ws_size;
    const float* left  = (const float*)d_in[0];
    const float* right = (const float*)d_in[1];
    float*       out   = (float*)d_out;

    constexpr int B = 4, C = 32, H = 64;
    dim3 grid(B * C * H);                    // 8192 blocks, one per (b,c,h) row
    dim3 block(256);                         // 8 wave32s
    cost_volume_kernel<<<grid, block, 0, stream>>>(left, right, out);
}